// GNN_35854386987741
// MI455X (gfx1250) — compile-verified
//
#include <hip/hip_runtime.h>
#include <hip/hip_bf16.h>

typedef __attribute__((ext_vector_type(2))) float v2f;
typedef __attribute__((ext_vector_type(8))) float v8f;

#define NNODES 100000
#define NEDGES 1600000
#define NFEAT_ 64
#define NHID_ 128
#define NCLASS_ 40

// -------- SpMM: out[row[e]] += w[e] * x[col[e]]  (segment-sum via L2 fp32 atomics)
__device__ __forceinline__ void atomic_add_f32(float* p, float v) {
    (void)__hip_atomic_fetch_add(p, v, __ATOMIC_RELAXED, __HIP_MEMORY_SCOPE_AGENT);
}

__global__ void spmm_atomic_kernel(const int* __restrict__ erow,
                                   const int* __restrict__ ecol,
                                   const float* __restrict__ ew,
                                   const float* __restrict__ x,
                                   float* __restrict__ out,
                                   int nedges, int log2q) {
    const long long idx = (long long)blockIdx.x * blockDim.x + threadIdx.x;
    const long long total = ((long long)nedges) << log2q;
    if (idx >= total) return;
    const int e = (int)(idx >> log2q);
    const int q = (int)(idx & ((1 << log2q) - 1));
    const int d = 4 << log2q;                 // feature width: 64 or 128
    const int src = ecol[e];
    const int dst = erow[e];
    const float w = ew[e];
    const float4 v = *(const float4*)(x + (long long)src * d + q * 4);
    float* o = out + (long long)dst * d + q * 4;
    atomic_add_f32(o + 0, w * v.x);
    atomic_add_f32(o + 1, w * v.y);
    atomic_add_f32(o + 2, w * v.z);
    atomic_add_f32(o + 3, w * v.w);
}

// -------- GC GEMM:  out[N,128] = (X*(1+eps) + Nb) @ W   using V_WMMA_F32_16X16X4_F32
// blockDim = 256 (8 waves); each wave owns one 16x16 output tile; grid.x = N/16.
__global__ void gc_gemm_kernel(const float* __restrict__ X,
                               const float* __restrict__ Nb,
                               const float* __restrict__ W,      // [K,128] row-major
                               const float* __restrict__ eps,
                               float* __restrict__ out,          // [N,128]
                               int K, int do_relu) {
    const int m0   = blockIdx.x * 16;
    const int wave = threadIdx.x >> 5;      // 0..7 -> column tile
    const int lane = threadIdx.x & 31;
    const int half = lane >> 4;             // 0: lanes 0-15, 1: lanes 16-31
    const int l16  = lane & 15;
    const int n0   = wave * 16;
    const float scale = 1.0f + eps[0];

    const int rowA = m0 + l16;              // N divisible by 16: always in range
    const float* xr = X  + rowA * K;
    const float* nr = Nb + rowA * K;

    v8f c = {};
    for (int k0 = 0; k0 < K; k0 += 4) {
        const int ka = k0 + 2 * half;       // ISA 16x4 f32 A layout: VGPR v holds K = 2*half + v
        v2f a, b;
        a.x = fmaf(xr[ka],     scale, nr[ka]);
        a.y = fmaf(xr[ka + 1], scale, nr[ka + 1]);
        b.x = W[ka * NHID_ + n0 + l16];
        b.y = W[(ka + 1) * NHID_ + n0 + l16];
        c = __builtin_amdgcn_wmma_f32_16x16x4_f32(false, a, false, b,
                                                  (short)0, c, false, false);
    }
#pragma unroll
    for (int r = 0; r < 8; ++r) {           // C/D: VGPR r = row r (lanes 0-15) / r+8 (16-31)
        const int row = m0 + r + 8 * half;
        float v = c[r];
        if (do_relu) v = fmaxf(v, 0.0f);
        out[row * NHID_ + n0 + l16] = v;
    }
}

// -------- FC head: logits[N,40] = Z @ fcW + b   (40 cols padded to 48 = 3 WMMA tiles)
// blockDim = 96 (3 waves); grid.x = N/16
__global__ void head_gemm_kernel(const float* __restrict__ Z,    // [N,128]
                                 const float* __restrict__ Wf,   // [128,40]
                                 const float* __restrict__ bias, // [40]
                                 float* __restrict__ out) {      // [N,40]
    const int m0   = blockIdx.x * 16;
    const int wave = threadIdx.x >> 5;
    const int lane = threadIdx.x & 31;
    const int half = lane >> 4;
    const int l16  = lane & 15;
    const int n0   = wave * 16;
    const int colB = n0 + l16;
    const bool colOK = (colB < NCLASS_);

    const float* zr = Z + (m0 + l16) * NHID_;

    v8f c = {};
    for (int k0 = 0; k0 < NHID_; k0 += 4) {
        const int ka = k0 + 2 * half;
        v2f a, b;
        a.x = zr[ka];
        a.y = zr[ka + 1];
        b.x = colOK ? Wf[ka * NCLASS_ + colB] : 0.0f;        // select, no divergence
        b.y = colOK ? Wf[(ka + 1) * NCLASS_ + colB] : 0.0f;
        c = __builtin_amdgcn_wmma_f32_16x16x4_f32(false, a, false, b,
                                                  (short)0, c, false, false);
    }
    if (colOK) {
        const float bb = bias[colB];
#pragma unroll
        for (int r = 0; r < 8; ++r) {
            const int row = m0 + r + 8 * half;
            out[row * NCLASS_ + colB] = c[r] + bb;
        }
    }
}

// -------- log_softmax over 40 classes, one wave32 per row, in place
__global__ void logsoftmax_kernel(float* __restrict__ out, int n) {
    const int warp = (int)((blockIdx.x * (long long)blockDim.x + threadIdx.x) >> 5);
    const int lane = threadIdx.x & 31;
    if (warp >= n) return;
    float* row = out + (long long)warp * NCLASS_;
    const float v0 = row[lane];                                   // lane < 32 < 40
    const float v1 = (lane < NCLASS_ - 32) ? row[32 + lane] : -3.402823466e38f;
    float m = fmaxf(v0, v1);
#pragma unroll
    for (int off = 16; off > 0; off >>= 1) m = fmaxf(m, __shfl_xor(m, off, 32));
    float s = __expf(v0 - m) + ((lane < NCLASS_ - 32) ? __expf(v1 - m) : 0.0f);
#pragma unroll
    for (int off = 16; off > 0; off >>= 1) s += __shfl_xor(s, off, 32);
    const float lse = m + __logf(s);
    row[lane] = v0 - lse;
    if (lane < NCLASS_ - 32) row[32 + lane] = v1 - lse;
}

extern "C" void kernel_launch(void* const* d_in, const int* in_sizes, int n_in,
                              void* d_out, int out_size, void* d_ws, size_t ws_size,
                              hipStream_t stream) {
    const float* x    = (const float*)d_in[0];
    const int*   erow = (const int*)  d_in[1];
    const int*   ecol = (const int*)  d_in[2];
    const float* ew   = (const float*)d_in[3];
    const float* W1   = (const float*)d_in[4];
    const float* eps1 = (const float*)d_in[5];
    const float* W2   = (const float*)d_in[6];
    const float* eps2 = (const float*)d_in[7];
    const float* fcW  = (const float*)d_in[8];
    const float* fcb  = (const float*)d_in[9];
    float* out = (float*)d_out;

    const size_t hidBytes = (size_t)NNODES * NHID_ * sizeof(float); // 51.2 MB
    char* ws = (char*)d_ws;
    float* h    = (float*)(ws);                  // [N,128]
    float* neib = (float*)(ws + hidBytes);       // [N,128] (layer1 uses first half)
    float* z    = (float*)(ws + 2 * hidBytes);   // [N,128]

    // ---- layer 1: spmm(x) -> neib ; (x*(1+e)+neib)@W1, relu -> h
    hipMemsetAsync(neib, 0, (size_t)NNODES * NFEAT_ * sizeof(float), stream);
    {
        const long long total = (long long)NEDGES * (NFEAT_ / 4);
        const int blocks = (int)((total + 255) / 256);
        spmm_atomic_kernel<<<blocks, 256, 0, stream>>>(erow, ecol, ew, x, neib, NEDGES, 4);
    }
    gc_gemm_kernel<<<NNODES / 16, 256, 0, stream>>>(x, neib, W1, eps1, h, NFEAT_, 1);

    // ---- layer 2: spmm(h) -> neib ; (h*(1+e)+neib)@W2 -> z
    hipMemsetAsync(neib, 0, hidBytes, stream);
    {
        const long long total = (long long)NEDGES * (NHID_ / 4);
        const int blocks = (int)((total + 255) / 256);
        spmm_atomic_kernel<<<blocks, 256, 0, stream>>>(erow, ecol, ew, h, neib, NEDGES, 5);
    }
    gc_gemm_kernel<<<NNODES / 16, 256, 0, stream>>>(h, neib, W2, eps2, z, NHID_, 0);

    // ---- head: z @ fcW + b -> out ; log_softmax in place
    head_gemm_kernel<<<NNODES / 16, 96, 0, stream>>>(z, fcW, fcb, out);
    logsoftmax_kernel<<<(NNODES + 7) / 8, 256, 0, stream>>>(out, NNODES);
}